// EdgeGCNLayer_39367670235775
// MI455X (gfx1250) — compile-verified
//
#include <hip/hip_runtime.h>
#include <hip/hip_bf16.h>
#include <cstdint>

#define B_    8
#define V_    10000
#define E_    160000
#define DIN   128
#define DOUT  128
#define ROWS  (B_ * V_)   // 80000 rows, = 5000 tiles of 16
#define KDIM  256
#define BNEPS 1e-5f

typedef __attribute__((ext_vector_type(2))) float v2f;
typedef __attribute__((ext_vector_type(8))) float v8f;

__device__ __forceinline__ void atomic_add_f32(float* p, float v) {
  __hip_atomic_fetch_add(p, v, __ATOMIC_RELAXED, __HIP_MEMORY_SCOPE_AGENT);
}

// ---------------- zero workspace (S + s_edge + sums + sumsq, contiguous) ----
__global__ void k_zero(float* __restrict__ p, int n4) {
  int i = blockIdx.x * blockDim.x + threadIdx.x;
  if (i < n4) ((float4*)p)[i] = make_float4(0.f, 0.f, 0.f, 0.f);
}

// ---------------- build Wt[k*128+o] = k<128 ? W_self[o,k] : W_node[o,k-128] -
__global__ void k_build_wt(const float* __restrict__ Wself,
                           const float* __restrict__ Wnode,
                           float* __restrict__ Wt) {
  int idx = blockIdx.x * 256 + threadIdx.x;     // 32768 total
  int k = idx >> 7, o = idx & 127;
  Wt[idx] = (k < DIN) ? Wself[o * DIN + k] : Wnode[o * DIN + (k - DIN)];
}

// ---------------- scatter X rows: S[b,dst,:] += X[b,src,:] -------------------
__global__ __launch_bounds__(256)
void k_scatter_x(const float* __restrict__ X, const int* __restrict__ eidx,
                 float* __restrict__ S) {
  const int e   = blockIdx.x;
  const int src = eidx[2 * e];
  const int dst = eidx[2 * e + 1];
  const int b   = threadIdx.x >> 5;
  const int c4  = threadIdx.x & 31;
  const float4 x = *(const float4*)(X + ((size_t)b * V_ + src) * DIN + c4 * 4);
  float* p = S + ((size_t)b * V_ + dst) * DIN + c4 * 4;
  atomic_add_f32(p + 0, x.x);
  atomic_add_f32(p + 1, x.y);
  atomic_add_f32(p + 2, x.z);
  atomic_add_f32(p + 3, x.w);
}

// ---------------- scatter edge_attr: s_edge[b,dst] += ea[b,e] ---------------
__global__ void k_scatter_e(const float* __restrict__ ea, const int* __restrict__ eidx,
                            float* __restrict__ sedge) {
  int e = blockIdx.x * 256 + threadIdx.x;
  int b = blockIdx.y;
  if (e < E_) {
    int dst = eidx[2 * e + 1];
    atomic_add_f32(sedge + (size_t)b * V_ + dst, ea[(size_t)b * E_ + e]);
  }
}

// ---------------- fused GEMM: H = [X|S] @ Wt + b_self + s_edge*W_edge -------
__global__ __launch_bounds__(256)
void k_gemm(const float* __restrict__ X, const float* __restrict__ S,
            const float* __restrict__ Wt, const float* __restrict__ bself,
            const float* __restrict__ Wedge, const float* __restrict__ sedge,
            float* __restrict__ H) {
  constexpr int LDA = 258;                 // bank-conflict-free stride
  __shared__ float As[16 * LDA];
  const int rtile = blockIdx.x;            // 0..4999
  const int tid   = threadIdx.x;

  // cooperative stage of A tile: 16 rows x 256 k (X cols 0..127, S cols 128..255)
  #pragma unroll
  for (int j = 0; j < 4; ++j) {
    int lin = tid + j * 256;               // 0..1023 float4 chunks
    int row = lin >> 6;
    int col = (lin & 63) * 4;
    int gr  = rtile * 16 + row;
    const float* sp = (col < DIN) ? (X + (size_t)gr * DIN + col)
                                  : (S + (size_t)gr * DIN + (col - DIN));
    float4 v = *(const float4*)sp;
    float* dp = &As[row * LDA + col];
    dp[0] = v.x; dp[1] = v.y; dp[2] = v.z; dp[3] = v.w;
  }
  __syncthreads();

  const int lane  = tid & 31;
  const int wave  = tid >> 5;              // column tile 0..7
  const int n     = lane & 15;             // M for A, N for B/C/D
  const int khalf = (lane >> 4) << 1;      // lanes 0-15 -> K+{0,1}; 16-31 -> K+{2,3}
  const int c16   = wave * 16;

  v8f acc = {};
  #pragma unroll 8
  for (int kk = 0; kk < 64; ++kk) {
    int k0 = kk * 4 + khalf;
    v2f a, b;
    a.x = As[n * LDA + k0];
    a.y = As[n * LDA + k0 + 1];
    b.x = Wt[(size_t)k0 * DOUT + c16 + n];
    b.y = Wt[(size_t)(k0 + 1) * DOUT + c16 + n];
    acc = __builtin_amdgcn_wmma_f32_16x16x4_f32(
        /*neg_a=*/false, a, /*neg_b=*/false, b,
        /*c_mod=*/(short)0, acc, /*reuse_a=*/false, /*reuse_b=*/false);
  }

  const int   colg = c16 + n;
  const float bias = bself[colg];
  const float we   = Wedge[colg];
  const int   rbase = rtile * 16 + ((lane < 16) ? 0 : 8);
  #pragma unroll
  for (int i = 0; i < 8; ++i) {
    int gr = rbase + i;
    float v = acc[i] + bias + sedge[gr] * we;
    H[(size_t)gr * DOUT + colg] = v;
  }
}

// ---------------- per-channel sum / sumsq over 80000 rows -------------------
__global__ __launch_bounds__(256)
void k_stats(const float* __restrict__ H, float* __restrict__ sums,
             float* __restrict__ sumsq) {
  __shared__ float ssum[128], ssq[128];
  const int tid = threadIdx.x;
  if (tid < 128) { ssum[tid] = 0.f; ssq[tid] = 0.f; }
  __syncthreads();
  const int c4 = tid & 31;
  const int rl = tid >> 5;
  float4 s = {0.f, 0.f, 0.f, 0.f}, q = {0.f, 0.f, 0.f, 0.f};
  for (int r = blockIdx.x * 8 + rl; r < ROWS; r += gridDim.x * 8) {
    float4 h = *(const float4*)(H + (size_t)r * DOUT + c4 * 4);
    s.x += h.x; s.y += h.y; s.z += h.z; s.w += h.w;
    q.x += h.x * h.x; q.y += h.y * h.y; q.z += h.z * h.z; q.w += h.w * h.w;
  }
  float sv[4] = {s.x, s.y, s.z, s.w}, qv[4] = {q.x, q.y, q.z, q.w};
  #pragma unroll
  for (int j = 0; j < 4; ++j) {
    __hip_atomic_fetch_add(&ssum[c4 * 4 + j], sv[j], __ATOMIC_RELAXED, __HIP_MEMORY_SCOPE_WORKGROUP);
    __hip_atomic_fetch_add(&ssq[c4 * 4 + j],  qv[j], __ATOMIC_RELAXED, __HIP_MEMORY_SCOPE_WORKGROUP);
  }
  __syncthreads();
  if (tid < 128) {
    atomic_add_f32(&sums[tid], ssum[tid]);
    atomic_add_f32(&sumsq[tid], ssq[tid]);
  }
}

// ---------------- batchnorm + relu, in place on H (== d_out) ----------------
__global__ void k_norm(float* __restrict__ H, const float* __restrict__ sums,
                       const float* __restrict__ sumsq, const float* __restrict__ gamma,
                       const float* __restrict__ beta) {
  const int idx = blockIdx.x * 256 + threadIdx.x;  // ROWS*32 exactly
  const int row = idx >> 5;
  const int c4  = idx & 31;
  const float invN = 1.0f / (float)ROWS;
  float* p = H + (size_t)row * DOUT + c4 * 4;
  float4 h = *(float4*)p;
  float hv[4] = {h.x, h.y, h.z, h.w}, ov[4];
  #pragma unroll
  for (int j = 0; j < 4; ++j) {
    int c = c4 * 4 + j;
    float mean  = sums[c] * invN;
    float var   = sumsq[c] * invN - mean * mean;
    float scale = gamma[c] * rsqrtf(var + BNEPS);
    float shift = beta[c] - mean * scale;
    float v = hv[j] * scale + shift;
    ov[j] = v > 0.f ? v : 0.f;
  }
  *(float4*)p = make_float4(ov[0], ov[1], ov[2], ov[3]);
}

extern "C" void kernel_launch(void* const* d_in, const int* in_sizes, int n_in,
                              void* d_out, int out_size, void* d_ws, size_t ws_size,
                              hipStream_t stream) {
  const float* X     = (const float*)d_in[0];
  const float* ea    = (const float*)d_in[1];
  const float* Wnode = (const float*)d_in[2];
  const float* Wedge = (const float*)d_in[3];
  const float* Wself = (const float*)d_in[4];
  const float* bself = (const float*)d_in[5];
  const float* gamma = (const float*)d_in[6];
  const float* beta  = (const float*)d_in[7];
  const int*   eidx  = (const int*)d_in[8];

  float* H  = (float*)d_out;                 // H buffer then final output (in place)
  float* ws = (float*)d_ws;
  float* S     = ws;                         // ROWS*DIN      = 10,240,000 f32
  float* sedge = S + (size_t)ROWS * DIN;     // ROWS          = 80,000
  float* sums  = sedge + ROWS;               // 128
  float* sumsq = sums + 128;                 // 128
  float* Wt    = sumsq + 128;                // KDIM*DOUT     = 32,768 (no zeroing needed)

  // zero S + sedge + sums + sumsq (contiguous): 10,320,256 floats
  const int zero_n4 = (ROWS * DIN + ROWS + 256) / 4;
  k_zero<<<(zero_n4 + 255) / 256, 256, 0, stream>>>(S, zero_n4);

  k_build_wt<<<(KDIM * DOUT) / 256, 256, 0, stream>>>(Wself, Wnode, Wt);

  k_scatter_x<<<E_, 256, 0, stream>>>(X, eidx, S);

  dim3 ge((E_ + 255) / 256, B_);
  k_scatter_e<<<ge, 256, 0, stream>>>(ea, eidx, sedge);

  k_gemm<<<ROWS / 16, 256, 0, stream>>>(X, S, Wt, bself, Wedge, sedge, H);

  k_stats<<<256, 256, 0, stream>>>(H, sums, sumsq);

  k_norm<<<(ROWS * 32) / 256, 256, 0, stream>>>(H, sums, sumsq, gamma, beta);
}